// _BilateralWeights_28896539968356
// MI455X (gfx1250) — compile-verified
//
#include <hip/hip_runtime.h>

// Problem constants (from reference): data[B,C,H,W], out[B,V,K,H,W]
#define B_ 16
#define C_ 5
#define H_ 64
#define W_ 2048
#define V_ 2
#define K_ 8

// Tiling: one block computes TH x TW output pixels for one batch b.
#define TH 8
#define TW 256
#define ROWS (TH + 2)          // staged rows incl. halo
#define LW   (TW + 8)          // staged cols: 4-elem (16B-aligned) halo each side
#define SEGS (LW / 4)          // 66 x b128 segments per staged row
#define NLOAD (C_ * ROWS * SEGS) // 3300 16-byte async copies per block

// Types matching the builtin's prototype: v4i in AS1 (global) / AS3 (LDS).
typedef int v4i __attribute__((vector_size(16)));
typedef __attribute__((address_space(1))) v4i* g_v4i_p;
typedef __attribute__((address_space(3))) v4i* l_v4i_p;

__device__ __forceinline__ void async_copy16(const float* gptr, float* lptr) {
#if __has_builtin(__builtin_amdgcn_global_load_async_to_lds_b128)
    // AS1 pointer values == 64-bit flat addresses for global memory.
    g_v4i_p gp = (g_v4i_p)(unsigned long long)gptr;
    // Low 32 bits of a flat LDS address are the LDS byte offset (ISA aperture
    // mapping) == the 32-bit AS3 pointer value.
    l_v4i_p lp = (l_v4i_p)(unsigned)(unsigned long long)lptr;
    __builtin_amdgcn_global_load_async_to_lds_b128(gp, lp, /*offset=*/0, /*cpol=*/0);
#else
    unsigned lds_off = (unsigned)(unsigned long long)lptr;
    unsigned long long gaddr = (unsigned long long)gptr;
    asm volatile("global_load_async_to_lds_b128 %0, %1, off"
                 :: "v"(lds_off), "v"(gaddr)
                 : "memory");
#endif
}

__device__ __forceinline__ void wait_async_zero() {
#if __has_builtin(__builtin_amdgcn_s_wait_asynccnt)
    __builtin_amdgcn_s_wait_asynccnt(0);
#else
    asm volatile("s_wait_asynccnt 0" ::: "memory");
#endif
}

__global__ __launch_bounds__(256)
void _BilateralWeights_28896539968356_kernel(const float* __restrict__ data,
                                             const float* __restrict__ sq_var,
                                             float* __restrict__ out) {
    __shared__ float sm[C_ * ROWS * LW];   // 5*10*264*4 = 52,800 B

    const int tid = threadIdx.x;
    const int w0  = blockIdx.x * TW;
    const int h0  = blockIdx.y * TH;
    const int b   = blockIdx.z;

    // ---- Stage (TH+2) x (TW+8) tile of all 5 channels into LDS via async DMA.
    // Addresses are clamped for safety; logically-OOB slots are never read
    // (border predicates substitute the zero-pad value in the compute phase).
    for (int i = tid; i < NLOAD; i += 256) {
        const int c   = i / (ROWS * SEGS);
        const int rem = i - c * (ROWS * SEGS);
        const int r   = rem / SEGS;
        const int seg = rem - r * SEGS;

        int row = h0 - 1 + r;
        row = row < 0 ? 0 : (row > H_ - 1 ? H_ - 1 : row);
        int col = w0 - 4 + seg * 4;                 // 16B aligned (w0 % 256 == 0)
        col = col < 0 ? 0 : (col > W_ - 4 ? W_ - 4 : col);

        const float* gp = data + (((size_t)(b * C_ + c) * H_ + row) * W_ + col);
        float*       lp = sm + ((c * ROWS + r) * LW + seg * 4);
        async_copy16(gp, lp);
    }
    wait_async_zero();
    __syncthreads();

    // ---- Compute: each thread owns column w for TH rows.
    const float ninv0 = -0.5f / sq_var[0];
    const float ninv1 = -0.5f / sq_var[1];

    const int w  = w0 + tid;
    const int lw = tid + 4;                // LDS column of the center pixel
    const bool lf = (w > 0);
    const bool rt = (w + 1 < W_);
    const size_t planeHW = (size_t)H_ * W_;

    // Neighbor offsets in reference (row-major, center excluded) order.
    constexpr int DI[K_] = {-1, -1, -1,  0, 0,  1, 1, 1};
    constexpr int DJ[K_] = {-1,  0,  1, -1, 1, -1, 0, 1};

    for (int lr = 0; lr < TH; ++lr) {
        const int  h  = h0 + lr;
        const bool up = (h > 0);
        const bool dn = (h + 1 < H_);

        float ctr[C_];
#pragma unroll
        for (int c = 0; c < C_; ++c)
            ctr[c] = sm[(c * ROWS + lr + 1) * LW + lw];

        float sq[K_];
#pragma unroll
        for (int k = 0; k < K_; ++k) {
            const int di = DI[k], dj = DJ[k];
            const bool rowOK = (di < 0) ? up : ((di > 0) ? dn : true);
            const bool colOK = (dj < 0) ? lf : ((dj > 0) ? rt : true);
            const bool valid = rowOK && colOK;
            float acc = 0.0f;
#pragma unroll
            for (int c = 0; c < C_; ++c) {
                const float n = valid
                    ? sm[(c * ROWS + lr + 1 + di) * LW + (lw + dj)]
                    : 0.0f;
                const float d = ctr[c] - n;
                acc = fmaf(d, d, acc);
            }
            sq[k] = acc;
        }

        // out[b][v][k][h][w]; write-once 128 MiB stream -> non-temporal stores.
        const size_t obase = (size_t)(b * V_ * K_) * planeHW + (size_t)h * W_ + w;
#pragma unroll
        for (int k = 0; k < K_; ++k) {
            const float s = sq[k];
            __builtin_nontemporal_store(__expf(s * ninv0),
                                        out + obase + (size_t)k * planeHW);
            __builtin_nontemporal_store(__expf(s * ninv1),
                                        out + obase + (size_t)(K_ + k) * planeHW);
        }
    }
}

extern "C" void kernel_launch(void* const* d_in, const int* in_sizes, int n_in,
                              void* d_out, int out_size, void* d_ws, size_t ws_size,
                              hipStream_t stream) {
    (void)in_sizes; (void)n_in; (void)out_size; (void)d_ws; (void)ws_size;
    const float* data   = (const float*)d_in[0];
    const float* sq_var = (const float*)d_in[1];
    float*       out    = (float*)d_out;

    dim3 grid(W_ / TW, H_ / TH, B_);   // 8 x 8 x 16 = 1024 blocks
    dim3 block(256);                    // 8 wave32s
    _BilateralWeights_28896539968356_kernel<<<grid, block, 0, stream>>>(data, sq_var, out);
}